// TokenSelector_53283364274703
// MI455X (gfx1250) — compile-verified
//
#include <hip/hip_runtime.h>
#include <hip/hip_bf16.h>

typedef float v2f __attribute__((ext_vector_type(2)));
typedef float v8f __attribute__((ext_vector_type(8)));

// Problem constants (from reference setup_inputs)
constexpr int kB    = 8;
constexpr int kS    = 16384;
constexpr int kD    = 512;
constexpr int kBlk  = 128;               // selection_block_size
constexpr int kNBlk = kS / kBlk;         // 128 blocks per batch row
constexpr int kTopK = 16;
constexpr int kNSel = kNBlk * kTopK;     // 2048 selected tokens per batch

#if __has_builtin(__builtin_amdgcn_wmma_f32_16x16x4_f32)
#define HAVE_WMMA_F32_16X16X4 1
#endif

// One workgroup (256 threads = 8 wave32) per (batch, block).
//   Phase 1: scores for 128 tokens via V_WMMA_F32_16X16X4_F32 (full f32 precision)
//   Phase 2: wave 0 does 16-round argmax top-k (ties -> lower index, sorted desc)
//   Phase 3: all waves gather the 16 selected rows (L2-hot) to the output
__global__ __launch_bounds__(256)
void TokenSelector_53283364274703_kernel(const float* __restrict__ x,
                                         const float* __restrict__ Wv,
                                         float* __restrict__ out) {
    const int tid  = threadIdx.x;
    const int lane = tid & 31;
    const int wv   = tid >> 5;                 // wave id 0..7
    const int batch = blockIdx.x >> 7;         // / kNBlk
    const int blk   = blockIdx.x & (kNBlk - 1);

    const float* __restrict__ xblk = x + ((size_t)batch * kS + (size_t)blk * kBlk) * kD;

    __shared__ float W_lds[kD];
    __shared__ float scores_lds[kBlk];
    __shared__ int   sel_lds[kTopK];

    // Stage W (512 f32) into LDS once: broadcast source for the B fragment.
    W_lds[tid]       = Wv[tid];
    W_lds[tid + 256] = Wv[tid + 256];
    __syncthreads();

    // ---------------- Phase 1: scoring ----------------
    // Wave wv owns tokens [16*wv, 16*wv+16).
    const int tok  = (wv << 4) + (lane & 15);   // M = lane % 16
    const int koff = (lane >> 4) << 1;          // K sub-offset: 0 (lanes 0-15) / 2 (16-31)

#if defined(HAVE_WMMA_F32_16X16X4)
    v8f acc = {};
    const float* __restrict__ arow = xblk + (size_t)tok * kD + koff;  // 8B aligned
    const float* __restrict__ wrow = &W_lds[koff];                    // 8B aligned
#pragma unroll 4
    for (int k0 = 0; k0 < kD; k0 += 4) {
        v2f a  = *(const v2f*)(arow + k0);   // A: x tile, 16x4 f32
        v2f bb = *(const v2f*)(wrow + k0);   // B: W replicated over all 16 columns
        acc = __builtin_amdgcn_wmma_f32_16x16x4_f32(
            /*neg_a=*/false, a, /*neg_b=*/false, bb,
            /*c_mod=*/(short)0, acc, /*reuse_a=*/false, /*reuse_b=*/false);
    }
    // C layout: VGPR r holds M = r + 8*(lane/16); every column identical.
    if ((lane & 15) == 0) {
        const int mbase = (wv << 4) + ((lane >> 4) << 3);
#pragma unroll
        for (int r = 0; r < 8; ++r) scores_lds[mbase + r] = acc[r];
    }
#else
    // Scalar fallback (same LDS result layout); only used if builtin missing.
    if ((lane & 15) == 0) {
        const int mbase = (wv << 4) + ((lane >> 4) << 3);
        for (int r = 0; r < 8; ++r) {
            const float* row = xblk + (size_t)(mbase + r) * kD;
            float s = 0.0f;
            for (int k = 0; k < kD; ++k) s += row[k] * W_lds[k];
            scores_lds[mbase + r] = s;
        }
    }
#endif
    __syncthreads();

    // ---------------- Phase 2: block top-k (wave 0) ----------------
    // Serially-dependent selection rounds: keep rolled (unroll 1) to avoid the
    // multi-thousand-instruction code blob the full unroll produced (I-cache).
    if (wv == 0) {
        float sv[4];
#pragma unroll
        for (int c = 0; c < 4; ++c) sv[c] = scores_lds[lane + (c << 5)];
        unsigned used = 0;
#pragma unroll 1
        for (int it = 0; it < kTopK; ++it) {
            float best = -3.402823466e38f;
            int   bi   = 0x7fffffff;
#pragma unroll
            for (int c = 0; c < 4; ++c) {
                if (!(used & (1u << c))) {
                    float v  = sv[c];
                    int   id = lane + (c << 5);
                    if (v > best || (v == best && id < bi)) { best = v; bi = id; }
                }
            }
            // full-wave argmax reduction, ties -> lower index (matches jax top_k)
#pragma unroll
            for (int off = 16; off >= 1; off >>= 1) {
                float ov = __shfl_xor(best, off, 32);
                int   oi = __shfl_xor(bi,   off, 32);
                if (ov > best || (ov == best && oi < bi)) { best = ov; bi = oi; }
            }
            if ((bi & 31) == lane) used |= 1u << (bi >> 5);  // retire winner
            if (lane == 0) sel_lds[it] = bi;                  // descending order
        }
    }
    __syncthreads();

    // ---------------- Phase 3: gather (L2-resident rows) ----------------
    float* __restrict__ outb = out + ((size_t)batch * kNSel + (size_t)blk * kTopK) * kD;
#pragma unroll
    for (int j = 0; j < 2; ++j) {
        const int slot = (wv << 1) + j;          // 8 waves x 2 rows = 16 rows
        const int t    = sel_lds[slot];
        const float4* __restrict__ src = (const float4*)(xblk + (size_t)t * kD);
        float4* __restrict__ dst       = (float4*)(outb + (size_t)slot * kD);
#pragma unroll
        for (int i = 0; i < 4; ++i) dst[lane + (i << 5)] = src[lane + (i << 5)];
    }
}

extern "C" void kernel_launch(void* const* d_in, const int* in_sizes, int n_in,
                              void* d_out, int out_size, void* d_ws, size_t ws_size,
                              hipStream_t stream) {
    const float* x  = (const float*)d_in[0];   // [8,16384,512] f32
    const float* Wv = (const float*)d_in[1];   // [1,512] f32
    // d_in[2] is the bias: a uniform additive shift cannot change top-k -> unused.
    float* out = (float*)d_out;                // [8,2048,512] f32

    dim3 grid(kB * kNBlk);   // 1024 workgroups, one per (batch, block)
    dim3 block(256);         // 8 wave32
    TokenSelector_53283364274703_kernel<<<grid, block, 0, stream>>>(x, Wv, out);
}